// get_loss_46377056862517
// MI455X (gfx1250) — compile-verified
//
#include <hip/hip_runtime.h>

#define P_            6
#define BATCH_        16384
#define KBINS         256
#define JGT           4
#define ROWS          (P_ * BATCH_)          /* 98304 */
#define WAVES_PER_BLK 8
#define NBLOCKS       (ROWS / WAVES_PER_BLK) /* 12288 */

typedef __attribute__((ext_vector_type(2))) float v2f;
typedef __attribute__((ext_vector_type(4))) float v4f;
typedef __attribute__((ext_vector_type(8))) float v8f;

// Exact f32 sum of all 32 lane values using V_WMMA_F32_16X16X4_F32 with B = ones.
// A layout (16x4 f32): lanes 0-15 VGPR0 -> A[m,0], lanes 16-31 VGPR0 -> A[m,2];
// VGPR1 (K=1,3) is zero.  D[m][n] = part[m] + part[m+16] for every column n.
// Summing the 8 C/D VGPRs per lane and adding the xor-16 partner yields the full
// 32-lane sum in every lane.  EXEC is all ones here (no divergence around call).
__device__ __forceinline__ float wave_sum32(float part) {
  v2f a; a[0] = part; a[1] = 0.0f;
  v2f bones; bones[0] = 1.0f; bones[1] = 1.0f;
  v8f c = {};
  c = __builtin_amdgcn_wmma_f32_16x16x4_f32(
      /*neg_a=*/false, a, /*neg_b=*/false, bones,
      /*c_mod=*/(short)0, c, /*reuse_a=*/false, /*reuse_b=*/false);
  float t = ((c[0] + c[1]) + (c[2] + c[3])) + ((c[4] + c[5]) + (c[6] + c[7]));
  t += __shfl_xor(t, 16, 32);
  return t;
}

__global__ __launch_bounds__(256) void loss_main(
    const float* __restrict__ pred_ind,   // [P, B, K]
    const float* __restrict__ pred_off,   // [P, B, K]
    const float* __restrict__ tparams,    // [B, P]
    const float* __restrict__ centers,    // [P, K]
    const int*   __restrict__ tidx,       // [B, P, J]
    float2*      __restrict__ partials)   // [NBLOCKS] {cls, reg}
{
  const int lane = threadIdx.x & 31;
  const int wv   = threadIdx.x >> 5;
  const int row  = blockIdx.x * WAVES_PER_BLK + wv;   // row in [0, ROWS)
  const int p    = row / BATCH_;
  const int b    = row - p * BATCH_;
  const size_t rbase = (size_t)row * KBINS;

  // ---- stream the 256-float row: 8 floats/lane, two non-temporal b128 loads
  const v4f* rp = (const v4f*)(pred_ind + rbase + (size_t)lane * 8);
  v4f x0 = __builtin_nontemporal_load(rp);
  v4f x1 = __builtin_nontemporal_load(rp + 1);

  // ---- row max (shfl tree; WMMA cannot do max)
  float m = fmaxf(fmaxf(fmaxf(x0[0], x0[1]), fmaxf(x0[2], x0[3])),
                  fmaxf(fmaxf(x1[0], x1[1]), fmaxf(x1[2], x1[3])));
#pragma unroll
  for (int i = 1; i < 32; i <<= 1) m = fmaxf(m, __shfl_xor(m, i, 32));

  // ---- sum exp(x - m): hardware v_exp_f32 (x-m <= 0, no range issues),
  //      lane-local 8-term sum, then WMMA wave reduction
  float s = ((__expf(x0[0] - m) + __expf(x0[1] - m)) + (__expf(x0[2] - m) + __expf(x0[3] - m)))
          + ((__expf(x1[0] - m) + __expf(x1[1] - m)) + (__expf(x1[2] - m) + __expf(x1[3] - m)));
  float S = wave_sum32(s);            // v_wmma_f32_16x16x4_f32 inside

  // ---- classification: logsumexp - x[label]; all lanes load the same label (uniform)
  const size_t ibase = ((size_t)b * P_ + p) * JGT;
  const int lab   = tidx[ibase];                         // j = 0
  const float xlab = pred_ind[rbase + lab];
  const float cls_c = (m + __logf(S) - xlab) * (1.0f / BATCH_);  // S in [1,256]

  // ---- regression: lanes 0..3 carry j=0..3 (others duplicate then zero; no divergence)
  const int jj   = lane & 3;
  const int idxj = tidx[ibase + jj];
  const float center = centers[p * KBINS + idxj];
  const float off    = pred_off[rbase + idxj];
  const float tp     = tparams[(size_t)b * P_ + p];
  const float diff   = tp - center - off;
  // params with p % (P/N_ACTIONS=2) == ROT_IDX=1 -> orientation loss; PERIOD_RATIO == 2.0f
  // |2*diff| is O(few): v_cos_f32 fast path is accurate here.
  float contrib = (p & 1) ? -__cosf(diff * 2.0f) : diff * diff;
  contrib = (lane < 4) ? contrib : 0.0f;
#pragma unroll
  for (int i = 1; i < 4; i <<= 1) contrib += __shfl_xor(contrib, i, 32);
  const float reg_c = contrib * (1.0f / (float)(BATCH_ * JGT)); // lane 0 holds j-sum

  // ---- per-block reduction over the 8 waves
  __shared__ float scls[WAVES_PER_BLK];
  __shared__ float sreg[WAVES_PER_BLK];
  if (lane == 0) { scls[wv] = cls_c; sreg[wv] = reg_c; }
  __syncthreads();
  if (threadIdx.x == 0) {
    float c = 0.0f, r = 0.0f;
#pragma unroll
    for (int i = 0; i < WAVES_PER_BLK; ++i) { c += scls[i]; r += sreg[i]; }
    partials[blockIdx.x] = make_float2(c, r);
  }
}

__global__ __launch_bounds__(256) void loss_final(
    const float2* __restrict__ partials, float* __restrict__ out)
{
  double c = 0.0, r = 0.0;
  for (int i = threadIdx.x; i < NBLOCKS; i += 256) {
    float2 v = partials[i];
    c += (double)v.x; r += (double)v.y;
  }
  __shared__ double sc[256];
  __shared__ double sr[256];
  sc[threadIdx.x] = c; sr[threadIdx.x] = r;
  __syncthreads();
  for (int st = 128; st > 0; st >>= 1) {
    if (threadIdx.x < st) {
      sc[threadIdx.x] += sc[threadIdx.x + st];
      sr[threadIdx.x] += sr[threadIdx.x + st];
    }
    __syncthreads();
  }
  if (threadIdx.x == 0) {
    out[0] = (float)sc[0];   // cls_loss
    out[1] = (float)sr[0];   // reg_loss
  }
}

extern "C" void kernel_launch(void* const* d_in, const int* in_sizes, int n_in,
                              void* d_out, int out_size, void* d_ws, size_t ws_size,
                              hipStream_t stream) {
  (void)in_sizes; (void)n_in; (void)out_size; (void)ws_size;
  const float* pred_ind = (const float*)d_in[0];
  const float* pred_off = (const float*)d_in[1];
  const float* tparams  = (const float*)d_in[2];
  const float* centers  = (const float*)d_in[3];
  const int*   tidx     = (const int*)d_in[4];
  float2* partials = (float2*)d_ws;   // 12288 * 8 B = 96 KB scratch

  loss_main<<<NBLOCKS, 256, 0, stream>>>(pred_ind, pred_off, tparams, centers,
                                         tidx, partials);
  loss_final<<<1, 256, 0, stream>>>(partials, (float*)d_out);
}